// DiffRenderer_68058051773070
// MI455X (gfx1250) — compile-verified
//
#include <hip/hip_runtime.h>

typedef float v2f __attribute__((ext_vector_type(2)));
typedef float v8f __attribute__((ext_vector_type(8)));

#define DGRID   96
#define NVOX    (96*96*96)
#define WIDTH   640
#define HEIGHT  480
#define HW      (WIDTH*HEIGHT)
#define NPTS    150000
#define NVIEWS  2
#define VOXEL   0.04f
#define TRUNC   5.0f
#define RAYINC  1.5f
#define DMINV   0.025f
#define NSTEPS  67
#define EPSF    1e-5f

// d_out layout (floats): [0]=depth_loss [1]=normal_loss, depths, depths_target, normals, normals_t
#define OFF_DEPTH 2
#define OFF_DT    (OFF_DEPTH + NVIEWS*HW)
#define OFF_N     (OFF_DT   + NVIEWS*HW)
#define OFF_NT    (OFF_N    + NVIEWS*HW*3)

// ---------------- init ----------------
__global__ void k_init(unsigned* occ32, float* g, float* gt, float* ngw, float* ntw,
                       unsigned* hm32, unsigned* scr) {
  const int NG = NVOX * 3;
  int stride = gridDim.x * blockDim.x;
  for (int j = blockIdx.x * blockDim.x + threadIdx.x; j < NG; j += stride) {
    ngw[j] = 0.f; ntw[j] = 0.f;
    if (j < NVOX / 4) occ32[j] = 0u;
    if (j < NVOX) { g[j] = TRUNC; gt[j] = TRUNC; }
    if (j < (4 * HW) / 4) hm32[j] = 0u;
    if (j < 16) {
      int k = j & 7;
      scr[j] = (k == 1 || k == 3) ? 0x7F800000u : 0u;  // +inf for min slots
    }
  }
}

// ---------------- scatter points ----------------
__global__ void k_scatter(const int* __restrict__ coords, const float* __restrict__ sdf,
                          const float* __restrict__ sdft,
                          unsigned char* occ, float* g, float* gt) {
  int i = blockIdx.x * blockDim.x + threadIdx.x;
  if (i >= NPTS) return;
  int l0 = coords[i * 4 + 3], l1 = coords[i * 4 + 2], l2 = coords[i * 4 + 1];
  int idx = (l0 * DGRID + l1) * DGRID + l2;
  occ[idx] = 1;
  g[idx]  = sdf[i];
  gt[idx] = sdft[i];
}

// ---------------- per-point world-frame normals scattered to grids ----------------
__device__ __forceinline__ float nval(const float* gr, const unsigned char* occ,
                                      int a, int b, int c) {
  a = min(max(a, 0), DGRID - 1); b = min(max(b, 0), DGRID - 1); c = min(max(c, 0), DGRID - 1);
  int id = (a * DGRID + b) * DGRID + c;
  return occ[id] ? gr[id] : 0.f;  // reference scatters into a zero grid for normals
}

__global__ void k_normals(const int* __restrict__ coords, const unsigned char* __restrict__ occ,
                          const float* __restrict__ g, const float* __restrict__ gt,
                          float* ngw, float* ntw) {
  int i = blockIdx.x * blockDim.x + threadIdx.x;
  if (i >= NPTS) return;
  int l0 = coords[i * 4 + 3], l1 = coords[i * 4 + 2], l2 = coords[i * 4 + 1];
  int idx = (l0 * DGRID + l1) * DGRID + l2;
  #pragma unroll
  for (int which = 0; which < 2; which++) {
    const float* gr = which ? gt : g;
    float* ng       = which ? ntw : ngw;
    float nx = nval(gr, occ, l0, l1, l2 + 1) - nval(gr, occ, l0, l1, l2 - 1);
    float ny = nval(gr, occ, l0, l1 + 1, l2) - nval(gr, occ, l0, l1 - 1, l2);
    float nz = nval(gr, occ, l0 + 1, l1, l2) - nval(gr, occ, l0 - 1, l1, l2);
    float len = sqrtf(nx * nx + ny * ny + nz * nz);
    float inv = -1.f / fmaxf(len, EPSF);
    ng[idx * 3 + 0] = nx * inv;
    ng[idx * 3 + 1] = ny * inv;
    ng[idx * 3 + 2] = nz * inv;
  }
}

// ---------------- raycast ----------------
__global__ __launch_bounds__(256)
void k_raycast(const float* __restrict__ grid, const unsigned char* __restrict__ occ,
               const float* __restrict__ ngw,
               const float* __restrict__ view, const float* __restrict__ K,
               const float* __restrict__ origin,
               float* depth_out, float* nimg_out, unsigned char* hm_out) {
  int p = blockIdx.x * blockDim.x + threadIdx.x;
  if (p >= HW) return;
  float fx = K[0], fy = K[5], cx = K[2], cy = K[6];
  // Rcw = view[:3,:3]^T  (cam->world)
  float r00 = view[0], r01 = view[4], r02 = view[8];
  float r10 = view[1], r11 = view[5], r12 = view[9];
  float r20 = view[2], r21 = view[6], r22 = view[10];
  float tx = view[3], ty = view[7], tz = view[11];
  float cwx = -(r00 * tx + r01 * ty + r02 * tz);
  float cwy = -(r10 * tx + r11 * ty + r12 * tz);
  float cwz = -(r20 * tx + r21 * ty + r22 * tz);
  float camx = (cwx - origin[0]) / VOXEL;
  float camy = (cwy - origin[1]) / VOXEL;
  float camz = (cwz - origin[2]) / VOXEL;
  int u = p % WIDTH, v = p / WIDTH;
  float dcx = ((float)u - cx) / fx, dcy = ((float)v - cy) / fy;
  float dwx = r00 * dcx + r01 * dcy + r02;
  float dwy = r10 * dcx + r11 * dcy + r12;
  float dwz = r20 * dcx + r21 * dcy + r22;
  float il = rsqrtf(dwx * dwx + dwy * dwy + dwz * dwz);
  dwx *= il; dwy *= il; dwz *= il;

  float prev_s = TRUNC;
  bool prev_ok = false, hit = false;
  float t_hit = 0.f;

  for (int k = 0; k < NSTEPS; k++) {
    float t = DMINV + RAYINC * (float)k;
    float px = camx + t * dwx, py = camy + t * dwy, pz = camz + t * dwz;

    // gfx1250 prefetch of next step's base voxel (global_prefetch_b8)
    {
      int n0 = min(max((int)floorf(px + RAYINC * dwx), 0), DGRID - 2);
      int n1 = min(max((int)floorf(py + RAYINC * dwy), 0), DGRID - 2);
      int n2 = min(max((int)floorf(pz + RAYINC * dwz), 0), DGRID - 2);
      __builtin_prefetch(&grid[(n0 * DGRID + n1) * DGRID + n2], 0, 1);
    }

    float f0x = floorf(px), f0y = floorf(py), f0z = floorf(pz);
    int c0 = (int)f0x, c1 = (int)f0y, c2 = (int)f0z;
    bool ok = (c0 >= 0) & (c0 <= DGRID - 2) & (c1 >= 0) & (c1 <= DGRID - 2) &
              (c2 >= 0) & (c2 <= DGRID - 2);
    c0 = min(max(c0, 0), DGRID - 2);
    c1 = min(max(c1, 0), DGRID - 2);
    c2 = min(max(c2, 0), DGRID - 2);
    float fx0 = px - f0x, fy0 = py - f0y, fz0 = pz - f0z;
    int base = (c0 * DGRID + c1) * DGRID + c2;

    float s = 0.f;
    bool okk = ok;
    #pragma unroll
    for (int dx = 0; dx < 2; dx++) {
      float wx = dx ? fx0 : 1.f - fx0;
      #pragma unroll
      for (int dy = 0; dy < 2; dy++) {
        float wy = dy ? fy0 : 1.f - fy0;
        #pragma unroll
        for (int dz = 0; dz < 2; dz++) {
          float wz = dz ? fz0 : 1.f - fz0;
          int id = base + dx * (DGRID * DGRID) + dy * DGRID + dz;
          s += wx * wy * wz * grid[id];
          okk = okk && (occ[id] != 0);
        }
      }
    }
    bool cross = prev_ok && okk && (prev_s > 0.f) && (s <= 0.f) && !hit;
    float denom = prev_s - s;
    float dd = (fabsf(denom) < 1e-8f) ? 1e-8f : denom;
    float frac = prev_s / dd;
    if (cross) t_hit = t - RAYINC + RAYINC * frac;
    hit = hit || cross;
    prev_s = s;
    prev_ok = okk;
    if (__all((int)hit)) break;
  }

  if (depth_out) depth_out[p] = hit ? t_hit * VOXEL : 0.f;
  float hx = camx + t_hit * dwx, hy = camy + t_hit * dwy, hz = camz + t_hit * dwz;
  int p0 = min(max((int)rintf(hx), 0), DGRID - 1);
  int p1 = min(max((int)rintf(hy), 0), DGRID - 1);
  int p2 = min(max((int)rintf(hz), 0), DGRID - 1);
  int id = (p0 * DGRID + p1) * DGRID + p2;
  float n0 = 0.f, n1 = 0.f, n2 = 0.f;
  if (hit) { n0 = ngw[id * 3 + 0]; n1 = ngw[id * 3 + 1]; n2 = ngw[id * 3 + 2]; }
  nimg_out[p * 3 + 0] = n0;  // world-frame; rotated in-place by WMMA pass
  nimg_out[p * 3 + 1] = n1;
  nimg_out[p * 3 + 2] = n2;
  hm_out[p] = hit ? 1 : 0;
}

// ---------------- in-place normal-image rotation via V_WMMA_F32_16X16X4_F32 ----------------
// D = A(16x4) * B(4x16): A rows = 16 pixels' world normals (padded), B = R^T padded.
// D[m][n] (n<3) = (R @ nw_m)[n] = camera-frame normal.
// 4 independent WMMAs per wave iteration (no D->A/B RAW, so they pipeline); all
// control flow around the WMMAs is wave-uniform so EXEC is all-ones (ISA 7.12).
__global__ __launch_bounds__(256)
void k_rotate_wmma(float* out, const float* __restrict__ view_matrix) {
  const int TILES_PER_IMG  = HW / 16;            // 19200
  const int GROUPS_PER_IMG = TILES_PER_IMG / 4;  // 4800 (groups never cross images)
  const int TOTAL_GROUPS   = 4 * GROUPS_PER_IMG; // 19200
  int lane  = threadIdx.x & 31;
  int wave  = (blockIdx.x * blockDim.x + threadIdx.x) >> 5;
  int nwav  = (gridDim.x * blockDim.x) >> 5;
  int ncol  = lane & 15;
  int half  = lane >> 4;

  for (int grp = wave; grp < TOTAL_GROUPS; grp += nwav) {
    int img = grp / GROUPS_PER_IMG;                      // wave-uniform
    int t0  = (grp - img * GROUPS_PER_IMG) * 4;          // first tile of group
    int vv  = img & 1;                                   // 0:n v0, 1:n v1, 2:nt v0, 3:nt v1
    float* base = out + ((img < 2) ? (OFF_N  + (size_t)img * HW * 3)
                                   : (OFF_NT + (size_t)(img - 2) * HW * 3));
    const float* V = view_matrix + vv * 16;              // R = view[:3,:3]

    // B: lanes 0-15 hold rows K=0 (b.x) / K=1 (b.y); lanes 16-31 rows K=2 / K=3
    int k0 = 2 * half, k1 = 2 * half + 1;
    v2f bm;
    bm.x = (ncol < 3 && k0 < 3) ? V[ncol * 4 + k0] : 0.f;  // B[k][n] = R[n][k]
    bm.y = (ncol < 3 && k1 < 3) ? V[ncol * 4 + k1] : 0.f;

    // Branchless A loads: every lane loads components (2*half) and 1 of its pixel
    // (both addresses always valid), then selects -- no EXEC divergence.
    v2f am[4];
    #pragma unroll
    for (int q = 0; q < 4; q++) {
      int pix = (t0 + q) * 16 + ncol;
      float a0 = base[pix * 3 + 2 * half];   // half=0 -> comp0, half=1 -> comp2
      float a1 = base[pix * 3 + 1];          // comp1 (ignored for half=1)
      am[q].x = a0;
      am[q].y = half ? 0.f : a1;
    }

    v8f c[4];
    #pragma unroll
    for (int q = 0; q < 4; q++) {
      v8f z = {};
      c[q] = __builtin_amdgcn_wmma_f32_16x16x4_f32(false, am[q], false, bm,
                                                   (short)0, z, false, false);
    }

    if (ncol < 3) {  // only N=0..2 columns are real components
      #pragma unroll
      for (int q = 0; q < 4; q++) {
        int rbase = (t0 + q) * 16 + half * 8;  // lanes 0-15: rows 0-7; 16-31: rows 8-15
        #pragma unroll
        for (int r = 0; r < 8; r++) base[(rbase + r) * 3 + ncol] = c[q][r];
      }
    }
  }
}

// ---------------- reductions ----------------
// scr per view v (8 u32 slots): [0]=cnt [1]=minD [2]=maxD [3]=minT [4]=maxT
//                               [5]=sum_l1_depth(f32) [6]=cnt_n [7]=sum_l1_norm(f32)
__global__ void k_reduce1(const float* __restrict__ dout, const float* __restrict__ dtall,
                          const unsigned char* __restrict__ hm, unsigned* scr) {
  __shared__ unsigned s_cnt[256], s_minD[256], s_maxD[256], s_minT[256], s_maxT[256], s_cn[256];
  __shared__ float s_sn[256];
  int v = blockIdx.y;
  int p = blockIdx.x * blockDim.x + threadIdx.x;
  int tid = threadIdx.x;

  bool hg = false, ht = false; float d = 0.f, dtv = 0.f; float sn = 0.f;
  if (p < HW) {
    hg  = hm[v * HW + p] != 0;
    ht  = hm[(2 + v) * HW + p] != 0;
    d   = dout[OFF_DEPTH + (size_t)v * HW + p];
    dtv = dtall[(size_t)v * HW + p];
  }
  bool m  = hg && (dtv != 0.f);
  bool mn = hg && ht;
  if (mn) {
    const float* n  = dout + OFF_N  + (size_t)v * HW * 3 + (size_t)p * 3;
    const float* nt = dout + OFF_NT + (size_t)v * HW * 3 + (size_t)p * 3;
    sn = fabsf(n[0] - nt[0]) + fabsf(n[1] - nt[1]) + fabsf(n[2] - nt[2]);
  }
  unsigned db = __float_as_uint(d), tb = __float_as_uint(dtv);  // non-negative -> order-preserving
  s_cnt[tid]  = m ? 1u : 0u;
  s_minD[tid] = m ? db : 0x7F800000u;
  s_maxD[tid] = m ? db : 0u;
  s_minT[tid] = m ? tb : 0x7F800000u;
  s_maxT[tid] = m ? tb : 0u;
  s_cn[tid]   = mn ? 1u : 0u;
  s_sn[tid]   = sn;
  __syncthreads();
  for (int off = 128; off > 0; off >>= 1) {
    if (tid < off) {
      s_cnt[tid] += s_cnt[tid + off];
      s_minD[tid] = min(s_minD[tid], s_minD[tid + off]);
      s_maxD[tid] = max(s_maxD[tid], s_maxD[tid + off]);
      s_minT[tid] = min(s_minT[tid], s_minT[tid + off]);
      s_maxT[tid] = max(s_maxT[tid], s_maxT[tid + off]);
      s_cn[tid]  += s_cn[tid + off];
      s_sn[tid]  += s_sn[tid + off];
    }
    __syncthreads();
  }
  if (tid == 0) {
    unsigned* b = scr + v * 8;
    atomicAdd(&b[0], s_cnt[0]);
    atomicMin(&b[1], s_minD[0]);
    atomicMax(&b[2], s_maxD[0]);
    atomicMin(&b[3], s_minT[0]);
    atomicMax(&b[4], s_maxT[0]);
    atomicAdd(&b[6], s_cn[0]);
    atomicAdd((float*)&b[7], s_sn[0]);
  }
}

__global__ void k_reduce2(const float* __restrict__ dout, const float* __restrict__ dtall,
                          const unsigned char* __restrict__ hm, unsigned* scr) {
  __shared__ float s_s[256];
  int v = blockIdx.y;
  int p = blockIdx.x * blockDim.x + threadIdx.x;
  int tid = threadIdx.x;
  const unsigned* b = scr + v * 8;
  unsigned cnt = b[0];
  float vminD = __uint_as_float(b[1]);
  float rngD  = __uint_as_float(b[2]) - vminD;
  float denD  = (rngD <= 0.f) ? 1.f : rngD;
  float vminT = __uint_as_float(b[3]);
  float rngT  = __uint_as_float(b[4]) - vminT;
  float denT  = (rngT <= 0.f) ? 1.f : rngT;
  float contrib = 0.f;
  if (p < HW && cnt > 0) {
    bool hg = hm[v * HW + p] != 0;
    float dtv = dtall[(size_t)v * HW + p];
    if (hg && dtv != 0.f) {
      float d = dout[OFF_DEPTH + (size_t)v * HW + p];
      contrib = fabsf((d - vminD) / denD - (dtv - vminT) / denT);
    }
  }
  s_s[tid] = contrib;
  __syncthreads();
  for (int off = 128; off > 0; off >>= 1) {
    if (tid < off) s_s[tid] += s_s[tid + off];
    __syncthreads();
  }
  if (tid == 0) atomicAdd((float*)&scr[v * 8 + 5], s_s[0]);
}

__global__ void k_final(const unsigned* __restrict__ scr, float* dout) {
  if (blockIdx.x == 0 && threadIdx.x == 0) {
    float dl = 0.f, nl = 0.f;
    #pragma unroll
    for (int v = 0; v < NVIEWS; v++) {
      const unsigned* b = scr + v * 8;
      unsigned cnt = b[0];
      float s = ((const float*)b)[5];
      if (cnt > 0) dl += (s / (float)cnt) * (1.0f / NVIEWS);
      unsigned cn = b[6];
      float snv = ((const float*)b)[7];
      if (cn > 0) nl += (snv / (float)(3u * cn)) * (1.0f / NVIEWS);
    }
    dout[0] = dl;
    dout[1] = nl;
  }
}

// ---------------- host wrapper ----------------
extern "C" void kernel_launch(void* const* d_in, const int* in_sizes, int n_in,
                              void* d_out, int out_size, void* d_ws, size_t ws_size,
                              hipStream_t stream) {
  (void)in_sizes; (void)n_in; (void)out_size; (void)ws_size;
  const int*   coords = (const int*)d_in[0];
  const float* origin = (const float*)d_in[1];
  const float* sdf    = (const float*)d_in[2];
  const float* sdft   = (const float*)d_in[3];
  const float* dts    = (const float*)d_in[4];
  const float* intr   = (const float*)d_in[5];
  const float* view   = (const float*)d_in[6];
  float* out = (float*)d_out;
  char*  ws  = (char*)d_ws;

  unsigned char* occ = (unsigned char*)ws;                    //   884,736 B
  float* g   = (float*)(ws + 884736);                         // 3,538,944 B
  float* gt  = (float*)(ws + 4423680);                        // 3,538,944 B
  float* ngw = (float*)(ws + 7962624);                        // 10,616,832 B
  float* ntw = (float*)(ws + 18579456);                       // 10,616,832 B
  unsigned char* hm = (unsigned char*)(ws + 29196288);        // 1,228,800 B
  unsigned* scr = (unsigned*)(ws + 30425088);                 // 64 B

  k_init<<<2048, 256, 0, stream>>>((unsigned*)occ, g, gt, ngw, ntw, (unsigned*)hm, scr);
  k_scatter<<<(NPTS + 255) / 256, 256, 0, stream>>>(coords, sdf, sdft, occ, g, gt);
  k_normals<<<(NPTS + 255) / 256, 256, 0, stream>>>(coords, occ, g, gt, ngw, ntw);

  for (int v = 0; v < NVIEWS; v++) {
    k_raycast<<<(HW + 255) / 256, 256, 0, stream>>>(
        g, occ, ngw, view + v * 16, intr + v * 16, origin,
        out + OFF_DEPTH + (size_t)v * HW, out + OFF_N + (size_t)v * HW * 3, hm + (size_t)v * HW);
    k_raycast<<<(HW + 255) / 256, 256, 0, stream>>>(
        gt, occ, ntw, view + v * 16, intr + v * 16, origin,
        nullptr, out + OFF_NT + (size_t)v * HW * 3, hm + (size_t)(2 + v) * HW);
  }

  hipMemcpyAsync(out + OFF_DT, dts, (size_t)NVIEWS * HW * sizeof(float),
                 hipMemcpyDeviceToDevice, stream);

  k_rotate_wmma<<<300, 256, 0, stream>>>(out, view);

  dim3 rgrid((HW + 255) / 256, NVIEWS);
  k_reduce1<<<rgrid, 256, 0, stream>>>(out, dts, hm, scr);
  k_reduce2<<<rgrid, 256, 0, stream>>>(out, dts, hm, scr);
  k_final<<<1, 32, 0, stream>>>(scr, out);
}